// MoBoAligner_83726092469097
// MI455X (gfx1250) — compile-verified
//
#include <hip/hip_runtime.h>
#include <hip/hip_bf16.h>
#include <math.h>

// Problem dims (fixed by the reference)
#define BB 8
#define TT 256
#define MM 1024
#define DD 512
#define TEMP_MIN 0.1f
#define TEMP_MAX 1.0f

typedef __attribute__((ext_vector_type(2))) float v2f;
typedef __attribute__((ext_vector_type(8))) float v8f;

// ---------------------------------------------------------------------------
// helpers
// ---------------------------------------------------------------------------
__device__ __forceinline__ float lae(float a, float b) {
    // logaddexp(a,b) = max(a,b) + log(1 + exp(-|a-b|))
    // 1 + exp(d) is in [1,2] so the fast v_log_f32 path is accurate here, and
    // avoids libm log1pf's denorm-fixup sequence in the 2e9-iteration hot loop.
    float mx = fmaxf(a, b);
    float d  = -fabsf(a - b);
    return mx + __logf(1.0f + __expf(d));
}

// ---------------------------------------------------------------------------
// K1: energy[b,t,m] = <text[b,t,:], mel[b,m,:]>  via V_WMMA_F32_16X16X4_F32,
//     then e = (energy + gumbel(u)) / temp  -> ws_e
// Block: 256 threads (8 waves). Block handles one (b, 16-row t-strip).
// Wave w computes m-tiles w, w+8, ..., each a 16x16 WMMA accumulation over D.
// ---------------------------------------------------------------------------
__global__ void __launch_bounds__(256)
k_energy(const float* __restrict__ text, const float* __restrict__ mel,
         const float* __restrict__ ratio, const float* __restrict__ u,
         float* __restrict__ e_out) {
    const int b  = blockIdx.x / (TT / 16);
    const int t0 = (blockIdx.x % (TT / 16)) * 16;

    const int wave   = threadIdx.x >> 5;
    const int lane   = threadIdx.x & 31;
    const int laneLo = lane & 15;
    const int hi     = lane >> 4;        // 0 for lanes 0-15, 1 for lanes 16-31

    const float* textB = text + (size_t)b * TT * DD;
    const float* melB  = mel  + (size_t)b * MM * DD;
    const float* uB    = u    + (size_t)b * TT * MM;
    float*       eB    = e_out + (size_t)b * TT * MM;

    const float invtemp = 1.0f / (TEMP_MIN + (TEMP_MAX - TEMP_MIN) * ratio[0]);
    const int  arow     = t0 + laneLo;   // A-matrix row (t) for this lane

    for (int mt = wave; mt < MM / 16; mt += 8) {
        const int m0   = mt * 16;
        const int mcol = m0 + laneLo;    // B-matrix column (m) for this lane

        // prefetch next m-tile's mel rows into cache (global_prefetch_b8)
        if (mt + 8 < MM / 16) {
            __builtin_prefetch(&melB[(size_t)(mcol + 128) * DD], 0, 1);
        }

        v8f c = {};
#pragma unroll 4
        for (int k0 = 0; k0 < DD; k0 += 4) {
            const int kk = k0 + (hi ? 2 : 0);
            // A: 16x4 fp32 tile of text   (lane=M row, VGPR0:K / VGPR1:K+1)
            v2f a;
            a.x = textB[(size_t)arow * DD + kk];
            a.y = textB[(size_t)arow * DD + kk + 1];
            // B: 4x16 fp32 tile of mel^T  (B[k][n] = mel[m0+n][k])
            v2f bb;
            bb.x = melB[(size_t)mcol * DD + kk];
            bb.y = melB[(size_t)mcol * DD + kk + 1];
            c = __builtin_amdgcn_wmma_f32_16x16x4_f32(
                    /*neg_a=*/false, a, /*neg_b=*/false, bb,
                    /*c_mod=*/(short)0, c, /*reuse_a=*/false, /*reuse_b=*/false);
        }

        // epilogue: add gumbel, scale by 1/temp, store scaled energy
#pragma unroll
        for (int r = 0; r < 8; ++r) {
            const int trow = t0 + r + hi * 8;
            const float uu = uB[(size_t)trow * MM + mcol];
            const float g  = -__logf(-__logf(uu));
            eB[(size_t)trow * MM + mcol] = (c[r] + g) * invtemp;
        }
    }
}

// ---------------------------------------------------------------------------
// K2: in-place log_softmax over m (row length M=1024). One wave per row.
// ---------------------------------------------------------------------------
__global__ void __launch_bounds__(256)
k_logsoftmax(float* __restrict__ e) {
    const int wave = threadIdx.x >> 5;
    const int lane = threadIdx.x & 31;
    const int row  = blockIdx.x * 8 + wave;      // flat (b*T + t), 2048 rows
    float* er = e + (size_t)row * MM;

    float mx = -INFINITY;
    for (int i = lane; i < MM; i += 32) mx = fmaxf(mx, er[i]);
#pragma unroll
    for (int off = 16; off > 0; off >>= 1) mx = fmaxf(mx, __shfl_xor(mx, off, 32));

    float sm = 0.0f;
    for (int i = lane; i < MM; i += 32) sm += __expf(er[i] - mx);
#pragma unroll
    for (int off = 16; off > 0; off >>= 1) sm += __shfl_xor(sm, off, 32);

    const float lz = mx + __logf(sm);
    for (int i = lane; i < MM; i += 32) er[i] = er[i] - lz;
}

// ---------------------------------------------------------------------------
// K3a: initialize alpha row buffer: row0[:-1] = [1, 0, ..., 0] per batch.
// ---------------------------------------------------------------------------
__global__ void __launch_bounds__(256)
k_alpha_init(float* __restrict__ row0) {
    const int idx = blockIdx.x * 256 + threadIdx.x;   // 0 .. B*M-1
    row0[idx] = ((idx % MM) == 0) ? 1.0f : 0.0f;
}

// ---------------------------------------------------------------------------
// K3b: one alpha step (launched T times; kernel boundary = global barrier).
//   s[m] = sum_j logaddexp(row_in[j], logp[b,ti,m]);  row_out = [0, s[0..M-2]]
// Grid: B*8 blocks x 128 threads (thread = one m) -> 64 WGPs per step.
// Row replicated into LDS per block, read as float4 (ds_load_b128).
// ---------------------------------------------------------------------------
__global__ void __launch_bounds__(128)
k_alpha_step(const float* __restrict__ logp,
             const float* __restrict__ row_in, float* __restrict__ row_out,
             float* __restrict__ alpha, int ti) {
    __shared__ __align__(16) float row_sh[MM];
    const int b  = blockIdx.x >> 3;
    const int m  = (blockIdx.x & 7) * 128 + threadIdx.x;

    const float* rin = row_in + (size_t)b * MM;
    for (int i = threadIdx.x; i < MM; i += 128) row_sh[i] = rin[i];
    __syncthreads();

    const float lp = logp[(size_t)b * TT * MM + (size_t)ti * MM + m];
    const float4* r4 = (const float4*)row_sh;
    float s = 0.0f;
#pragma unroll 4
    for (int j = 0; j < MM / 4; ++j) {
        const float4 rv = r4[j];
        s += lae(rv.x, lp);
        s += lae(rv.y, lp);
        s += lae(rv.z, lp);
        s += lae(rv.w, lp);
    }

    alpha[(size_t)b * TT * MM + (size_t)ti * MM + m] = s;

    float* rout = row_out + (size_t)b * MM;
    if (m < MM - 1) rout[m + 1] = s;      // next row = [0, s[0..M-2]]
    if (m == 0)     rout[0]     = 0.0f;   // padded column 0
}

// ---------------------------------------------------------------------------
// K4: beta recursion. Row is a broadcast scalar after the first step, so we
// only store beta_val[b, i] for i < T-1. One 256-thread block per b.
// ---------------------------------------------------------------------------
__global__ void __launch_bounds__(256)
k_beta(const float* __restrict__ logp, float* __restrict__ beta_val) {
    __shared__ float red[256];
    const int b   = blockIdx.x;
    const int tid = threadIdx.x;
    const float* lpB = logp + (size_t)b * TT * MM;

    float v     = 0.0f;   // broadcast row value (valid after first step)
    bool  first = true;

    for (int i = TT - 2; i >= 0; --i) {
        float partial = 0.0f;
#pragma unroll
        for (int q = 0; q < 4; ++q) {
            const int m   = tid + q * 256;
            const float r = first ? ((m == MM - 1) ? 1.0f : 0.0f) : v;
            partial += lae(r, lpB[(size_t)i * MM + m]);
        }
        red[tid] = partial;
        __syncthreads();
        for (int s2 = 128; s2 > 0; s2 >>= 1) {
            if (tid < s2) red[tid] += red[tid + s2];
            __syncthreads();
        }
        v = red[0];
        __syncthreads();
        if (tid == 0) beta_val[b * TT + i] = v;
        first = false;
    }
}

// ---------------------------------------------------------------------------
// K5: gamma = alpha + beta (in place over alpha buffer);
//     gamma_log = gamma - logsumexp_t(gamma)  -> d_out[0 .. B*T*M)
// Thread = one (b, m) column; loops over T. Coalesced across m.
// ---------------------------------------------------------------------------
__global__ void __launch_bounds__(256)
k_gamma(float* __restrict__ alpha, const float* __restrict__ beta_val,
        float* __restrict__ gamma_log) {
    const int idx = blockIdx.x * 256 + threadIdx.x;   // 0 .. B*M-1
    const int b = idx / MM;
    const int m = idx % MM;
    float* aB = alpha + (size_t)b * TT * MM;

    // pass 1: online logsumexp over t
    float mx = -INFINITY, sm = 0.0f;
    for (int t = 0; t < TT; ++t) {
        const float bt = (t == TT - 1) ? ((m == MM - 1) ? 1.0f : 0.0f)
                                       : beta_val[b * TT + t];
        const float g = aB[(size_t)t * MM + m] + bt;
        if (g > mx) { sm = sm * __expf(mx - g) + 1.0f; mx = g; }
        else        { sm += __expf(g - mx); }
    }
    const float lse = mx + __logf(sm);

    // pass 2: write gamma (in place) and gamma_log
    for (int t = 0; t < TT; ++t) {
        const float bt = (t == TT - 1) ? ((m == MM - 1) ? 1.0f : 0.0f)
                                       : beta_val[b * TT + t];
        const float g = aB[(size_t)t * MM + m] + bt;
        aB[(size_t)t * MM + m] = g;
        gamma_log[(size_t)b * TT * MM + (size_t)t * MM + m] = g - lse;
    }
}

// ---------------------------------------------------------------------------
// K6: expanded[b,m,d] = sum_t gamma[b,t,m] * text[b,t,d]  via fp32 WMMA.
// A = gamma^T tile (16 m-rows x 4 t), B = text tile (4 t x 16 d). K loop = T.
// Block: 256 threads / 8 waves; block = one (b, 16-row m-tile); waves split D.
// ---------------------------------------------------------------------------
__global__ void __launch_bounds__(256)
k_expand(const float* __restrict__ gamma, const float* __restrict__ text,
         float* __restrict__ out) {
    const int b  = blockIdx.x / (MM / 16);
    const int m0 = (blockIdx.x % (MM / 16)) * 16;

    const int wave   = threadIdx.x >> 5;
    const int lane   = threadIdx.x & 31;
    const int laneLo = lane & 15;
    const int hi     = lane >> 4;

    const float* gB = gamma + (size_t)b * TT * MM;
    const float* tB = text  + (size_t)b * TT * DD;
    float*       oB = out   + (size_t)b * MM * DD;

    const int mrow = m0 + laneLo;

    for (int nt = wave; nt < DD / 16; nt += 8) {
        const int d0   = nt * 16;
        const int ncol = d0 + laneLo;

        v8f c = {};
#pragma unroll 4
        for (int k0 = 0; k0 < TT; k0 += 4) {
            const int kt = k0 + (hi ? 2 : 0);
            v2f a;   // A[mrow][kt] = gamma[b, kt, mrow]  (transposed read)
            a.x = gB[(size_t)kt * MM + mrow];
            a.y = gB[(size_t)(kt + 1) * MM + mrow];
            v2f bb;  // B[kt][ncol] = text[b, kt, ncol]
            bb.x = tB[(size_t)kt * DD + ncol];
            bb.y = tB[(size_t)(kt + 1) * DD + ncol];
            c = __builtin_amdgcn_wmma_f32_16x16x4_f32(
                    false, a, false, bb, (short)0, c, false, false);
        }
#pragma unroll
        for (int r = 0; r < 8; ++r) {
            const int m = m0 + r + hi * 8;
            oB[(size_t)m * DD + ncol] = c[r];
        }
    }
}

// ---------------------------------------------------------------------------
// launch
// ---------------------------------------------------------------------------
extern "C" void kernel_launch(void* const* d_in, const int* in_sizes, int n_in,
                              void* d_out, int out_size, void* d_ws, size_t ws_size,
                              hipStream_t stream) {
    const float* text  = (const float*)d_in[0];   // [B,T,D]
    const float* mel   = (const float*)d_in[1];   // [B,M,D]
    const float* ratio = (const float*)d_in[2];   // scalar
    const float* noise = (const float*)d_in[3];   // [B,T,M]

    float* out = (float*)d_out;                   // gamma_log ++ expanded

    float* ws       = (float*)d_ws;
    float* e_logp   = ws;                                   // B*T*M (energy -> logp)
    float* alphabuf = ws + (size_t)BB * TT * MM;            // B*T*M (alpha -> gamma)
    float* betabuf  = ws + (size_t)2 * BB * TT * MM;        // B*T
    float* rowbuf   = betabuf + BB * TT;                    // 2 * B*M (double buffer)

    float* gamma_log = out;
    float* expanded  = out + (size_t)BB * TT * MM;

    // 1. energy + gumbel + temperature (WMMA f32)
    k_energy<<<dim3(BB * (TT / 16)), dim3(256), 0, stream>>>(text, mel, ratio,
                                                             noise, e_logp);
    // 2. log_softmax over m (in place)
    k_logsoftmax<<<dim3(BB * TT / 8), dim3(256), 0, stream>>>(e_logp);
    // 3. alpha recursion: T step-kernels, double-buffered row (global barrier
    //    per step via kernel boundary; 64 WGPs busy per step)
    k_alpha_init<<<dim3(BB * MM / 256), dim3(256), 0, stream>>>(rowbuf);
    for (int t = 1; t <= TT; ++t) {
        const float* rin  = rowbuf + (size_t)((t - 1) & 1) * BB * MM;
        float*       rout = rowbuf + (size_t)(t & 1) * BB * MM;
        k_alpha_step<<<dim3(BB * 8), dim3(128), 0, stream>>>(e_logp, rin, rout,
                                                             alphabuf, t - 1);
    }
    // 4. beta recursion (broadcast-scalar rows)
    k_beta<<<dim3(BB), dim3(256), 0, stream>>>(e_logp, betabuf);
    // 5. gamma + logsumexp over t -> gamma_log; gamma overwrites alpha buffer
    k_gamma<<<dim3(BB * MM / 256), dim3(256), 0, stream>>>(alphabuf, betabuf,
                                                           gamma_log);
    // 6. expanded = gamma x text (WMMA f32)
    k_expand<<<dim3(BB * (MM / 16)), dim3(256), 0, stream>>>(alphabuf, text,
                                                             expanded);
}